// KernelModule51_71768903516180
// MI455X (gfx1250) — compile-verified
//
#include <hip/hip_runtime.h>
#include <math.h>

// 2-D inverse FFT (4096x4096) via four-step radix-64 decomposition.
// Complex 64x64x64 matmuls run on V_WMMA_F32_16X16X4_F32 (fp32 tensor op,
// needed for fp32-reference accuracy). Bandwidth-bound pipeline:
//   init tables -> prep(+ifftshift) -> rowIFFT -> transpose(re,im)
//   -> rowIFFT -> transposeRe
// Workspace: 4 planes of 4096^2 fp32 (256 MB) + 64 KB of tables in d_ws.

#define FFT_N   4096
#define FFT_NN  (FFT_N * FFT_N)
#define LDP     66                       // padded LDS row stride (floats)
#define PI2F    6.28318530717958647692f

typedef __attribute__((ext_vector_type(2))) float v2f;
typedef __attribute__((ext_vector_type(8))) float v8f;

__device__ __forceinline__ v8f wmma_f32_4(v2f a, v2f b, v8f c) {
  // D = A(16x4) * B(4x16) + C, fp32 everywhere.
  return __builtin_amdgcn_wmma_f32_16x16x4_f32(false, a, false, b,
                                               (short)0, c, false, false);
}

// ---- fragment helpers (ISA 7.12.2 layouts) ---------------------------------
// A-matrix frag (16x4): lane L: m = L&15, k = 2*(L>>4) + {0,1}
__device__ __forceinline__ v2f ldsA(const float* M, int mbase, int kbase, int lane) {
  int m  = mbase + (lane & 15);
  int kb = kbase + ((lane >> 4) << 1);
  v2f a; a.x = M[m * LDP + kb]; a.y = M[m * LDP + kb + 1]; return a;
}
// B-matrix frag (4x16): lane L: n = L&15, k = 2*(L>>4) + {0,1}
__device__ __forceinline__ v2f ldsB(const float* M, int kbase, int nbase, int lane) {
  int n  = nbase + (lane & 15);
  int kb = kbase + ((lane >> 4) << 1);
  v2f b; b.x = M[kb * LDP + n]; b.y = M[(kb + 1) * LDP + n]; return b;
}

// ---- one-time table generation ---------------------------------------------
// tab[0..4095]      : Re F64[j][k] = cos(2*pi*j*k/64)/64    (i = j*64+k)
// tab[4096..8191]   : Im F64[j][k] = sin(2*pi*j*k/64)/64
// tab[8192..12287]  : cos(2*pi*t/4096)
// tab[12288..16383] : sin(2*pi*t/4096)
__global__ __launch_bounds__(256) void init_tables(float* __restrict__ tab) {
  int i = blockIdx.x * 256 + threadIdx.x;     // 0..4095
  int j = i >> 6, k = i & 63;
  float s, c;
  sincosf((float)((j * k) & 63) * (PI2F / 64.0f), &s, &c);
  tab[i]          = c * (1.0f / 64.0f);
  tab[4096 + i]   = s * (1.0f / 64.0f);
  sincosf((float)i * (PI2F / (float)FFT_N), &s, &c);
  tab[8192 + i]   = c;
  tab[12288 + i]  = s;
}

// ---- pass 0: elementwise + ifftshift ---------------------------------------
// ifftshift on even 4096x4096 == XOR flat index with (2048<<12)|2048.
__global__ __launch_bounds__(256) void prep_kernel(
    const float* __restrict__ xr, const float* __restrict__ xi,
    const float* __restrict__ mk, const float* __restrict__ pk,
    float* __restrict__ ar, float* __restrict__ ai) {
  int idx = blockIdx.x * 256 + threadIdx.x;
  int src = idx ^ ((2048 << 12) | 2048);
  float r = xr[src], im = xi[src];
  float mag = sqrtf(r * r + im * im) * mk[src];
  float ph  = atan2f(im, r) * pk[src];
  float s, c;
  sincosf(ph, &s, &c);
  ar[idx] = mag * c;
  ai[idx] = mag * s;
}

// ---- pass 1/2: batched 4096-point inverse FFT along rows -------------------
// One workgroup (8 wave32) per row. Row viewed as 64x64 (k1 slow, k2 fast):
//   G = F * X          (complex 64x64x64 matmul, WMMA)
//   B = G o twiddle    (w = exp(+2*pi*i*n1*k2/4096), LDS table lookup)
//   C = B * F          (complex 64x64x64 matmul, WMMA)
//   out[n1 + 64*n2] = C[n1][n2]   (transpose staged through LDS)
__global__ __launch_bounds__(256) void fft4096_rows(
    const float* __restrict__ inr, const float* __restrict__ ini,
    float* __restrict__ outr, float* __restrict__ outi,
    const float* __restrict__ tab) {
  extern __shared__ float smem[];
  float* sDr = smem;                   // 64*LDP data (re)
  float* sDi = sDr + 64 * LDP;         // 64*LDP data (im)
  float* sFr = sDi + 64 * LDP;         // 64*LDP F matrix (re)
  float* sFi = sFr + 64 * LDP;         // 64*LDP F matrix (im)
  float* sWc = sFi + 64 * LDP;         // 4096 twiddle cos
  float* sWs = sWc + 4096;             // 4096 twiddle sin

  const int tid  = threadIdx.x;
  const int lane = tid & 31;
  const int wid  = tid >> 5;
  const long rowBase = (long)blockIdx.x * FFT_N;

  // Stream row data + constant tables into LDS (coalesced, L2-resident tables).
  for (int i = tid; i < 64 * 64; i += 256) {
    int j = i >> 6, k = i & 63;
    sDr[j * LDP + k] = inr[rowBase + i];
    sDi[j * LDP + k] = ini[rowBase + i];
    sFr[j * LDP + k] = tab[i];
    sFi[j * LDP + k] = tab[4096 + i];
    sWc[i] = tab[8192 + i];
    sWs[i] = tab[12288 + i];
  }
  __syncthreads();

  // Stage 1: G = F * X, then twiddle. 16 output tiles, 2 per wave.
  v8f g_r[2], g_i[2];
#pragma unroll
  for (int t = 0; t < 2; ++t) {
    const int tile = wid + t * 8;
    const int mb = (tile >> 2) * 16, nb = (tile & 3) * 16;
    v8f rr = {}, ri = {}, ir = {}, ii = {};
#pragma unroll
    for (int kt = 0; kt < 16; ++kt) {
      const int kb = kt * 4;
      v2f aR = ldsA(sFr, mb, kb, lane);
      v2f aI = ldsA(sFi, mb, kb, lane);
      v2f bR = ldsB(sDr, kb, nb, lane);
      v2f bI = ldsB(sDi, kb, nb, lane);
      rr = wmma_f32_4(aR, bR, rr);
      ii = wmma_f32_4(aI, bI, ii);
      ri = wmma_f32_4(aR, bI, ri);
      ir = wmma_f32_4(aI, bR, ir);
    }
    v8f gr = rr - ii, gi = ri + ir;
#pragma unroll
    for (int e = 0; e < 8; ++e) {
      const int n1 = mb + e + ((lane >> 4) << 3);
      const int k2 = nb + (lane & 15);
      const int tw = (n1 * k2) & (FFT_N - 1);
      const float c = sWc[tw], s = sWs[tw];
      float tr = gr[e] * c - gi[e] * s;
      float ti = gr[e] * s + gi[e] * c;
      gr[e] = tr; gi[e] = ti;
    }
    g_r[t] = gr; g_i[t] = gi;
  }
  __syncthreads();  // all waves done reading X from LDS

  // Write twiddled G back into LDS.
#pragma unroll
  for (int t = 0; t < 2; ++t) {
    const int tile = wid + t * 8;
    const int mb = (tile >> 2) * 16, nb = (tile & 3) * 16;
#pragma unroll
    for (int e = 0; e < 8; ++e) {
      const int m = mb + e + ((lane >> 4) << 3);
      const int n = nb + (lane & 15);
      sDr[m * LDP + n] = g_r[t][e];
      sDi[m * LDP + n] = g_i[t][e];
    }
  }
  __syncthreads();

  // Stage 2: C = G * F.
  v8f c_r[2], c_i[2];
#pragma unroll
  for (int t = 0; t < 2; ++t) {
    const int tile = wid + t * 8;
    const int mb = (tile >> 2) * 16, nb = (tile & 3) * 16;
    v8f rr = {}, ri = {}, ir = {}, ii = {};
#pragma unroll
    for (int kt = 0; kt < 16; ++kt) {
      const int kb = kt * 4;
      v2f aR = ldsA(sDr, mb, kb, lane);
      v2f aI = ldsA(sDi, mb, kb, lane);
      v2f bR = ldsB(sFr, kb, nb, lane);
      v2f bI = ldsB(sFi, kb, nb, lane);
      rr = wmma_f32_4(aR, bR, rr);
      ii = wmma_f32_4(aI, bI, ii);
      ri = wmma_f32_4(aR, bI, ri);
      ir = wmma_f32_4(aI, bR, ir);
    }
    c_r[t] = rr - ii;
    c_i[t] = ri + ir;
  }
  __syncthreads();  // all waves done reading G from LDS

  // Four-step output transpose: stage C^T in LDS so global write coalesces.
#pragma unroll
  for (int t = 0; t < 2; ++t) {
    const int tile = wid + t * 8;
    const int mb = (tile >> 2) * 16, nb = (tile & 3) * 16;
#pragma unroll
    for (int e = 0; e < 8; ++e) {
      const int n1 = mb + e + ((lane >> 4) << 3);
      const int n2 = nb + (lane & 15);
      sDr[n2 * LDP + n1] = c_r[t][e];
      sDi[n2 * LDP + n1] = c_i[t][e];
    }
  }
  __syncthreads();

  for (int i = tid; i < 64 * 64; i += 256) {
    outr[rowBase + i] = sDr[(i >> 6) * LDP + (i & 63)];
    outi[rowBase + i] = sDi[(i >> 6) * LDP + (i & 63)];
  }
}

// ---- tiled transpose (coalesced both sides) --------------------------------
__global__ __launch_bounds__(256) void transpose32(const float* __restrict__ in,
                                                   float* __restrict__ out) {
  __shared__ float tile[32][33];
  const int bx = blockIdx.x * 32, by = blockIdx.y * 32;
  const int x = bx + threadIdx.x;
  for (int i = threadIdx.y; i < 32; i += 8)
    tile[i][threadIdx.x] = in[(long)(by + i) * FFT_N + x];
  __syncthreads();
  const int x2 = by + threadIdx.x;
  for (int i = threadIdx.y; i < 32; i += 8)
    out[(long)(bx + i) * FFT_N + x2] = tile[threadIdx.x][i];
}

// ---- driver ----------------------------------------------------------------
extern "C" void kernel_launch(void* const* d_in, const int* in_sizes, int n_in,
                              void* d_out, int out_size, void* d_ws, size_t ws_size,
                              hipStream_t stream) {
  const float* xr = (const float*)d_in[0];
  const float* xi = (const float*)d_in[1];
  const float* mk = (const float*)d_in[2];
  const float* pk = (const float*)d_in[3];
  float* out = (float*)d_out;

  // Workspace layout: 4 planes of FFT_NN floats (256 MB) + 16384-float tables.
  float* Ar  = (float*)d_ws;
  float* Ai  = Ar + (long)FFT_NN;
  float* Br  = Ar + 2L * FFT_NN;
  float* Bi  = Ar + 3L * FFT_NN;
  float* tab = Ar + 4L * FFT_NN;

  dim3 tgrid(FFT_N / 32, FFT_N / 32), tblk(32, 8);
  const size_t shbytes = (size_t)(4 * 64 * LDP + 2 * 4096) * sizeof(float);

  // DFT-matrix + twiddle tables (64 KB, L2-resident afterwards).
  init_tables<<<16, 256, 0, stream>>>(tab);
  // combined = mag*exp(i*ph), ifftshift folded into indexing.
  prep_kernel<<<FFT_NN / 256, 256, 0, stream>>>(xr, xi, mk, pk, Ar, Ai);
  // IFFT along rows.
  fft4096_rows<<<FFT_N, 256, shbytes, stream>>>(Ar, Ai, Br, Bi, tab);
  // Transpose both planes.
  transpose32<<<tgrid, tblk, 0, stream>>>(Br, Ar);
  transpose32<<<tgrid, tblk, 0, stream>>>(Bi, Ai);
  // IFFT along (former) columns.
  fft4096_rows<<<FFT_N, 256, shbytes, stream>>>(Ar, Ai, Br, Bi, tab);
  // Real part, transposed back into final orientation.
  transpose32<<<tgrid, tblk, 0, stream>>>(Br, out);
}